// TileRenderer_22539988369478
// MI455X (gfx1250) — compile-verified
//
#include <hip/hip_runtime.h>
#include <math.h>

// TileRenderer: bilinear-shift 4 source images per ptile and sum.
//   locs:    (1024, 4, 2)            f32
//   sources: (1024, 4, 5, 52, 52)    f32
//   out:     (1024, 5, 52, 52)       f32
// Memory bound: ~277 MB / 23.3 TB/s ~= 12 us. Strategy: one block per
// (ptile, band); async-copy the four 52x52 tiles into LDS (CDNA5
// GLOBAL_LOAD_ASYNC_TO_LDS_B128, ASYNCcnt), then do all bilinear gather
// taps from LDS.

#define PSLEN 52
#define NPIX  (PSLEN * PSLEN)   // 2704
#define NSRC  4
#define NBAND 5
#define BLOCK 256

// __builtin_amdgcn_global_load_async_to_lds_b128 signature:
//   (int4 addrspace(1)*, int4 addrspace(3)*, imm offset, imm cpol)
typedef int v4i __attribute__((ext_vector_type(4)));
typedef __attribute__((address_space(1))) v4i global_v4i;
typedef __attribute__((address_space(3))) v4i lds_v4i;

__device__ __forceinline__ float corner_tap(const float* __restrict__ tile,
                                            float iyf, float ixf, float wgt) {
    // Matches reference: mask computed on unclamped coords, gather on clamped.
    const bool valid = (ixf >= 0.0f) & (ixf <= (float)(PSLEN - 1)) &
                       (iyf >= 0.0f) & (iyf <= (float)(PSLEN - 1));
    const int ixc = (int)fminf(fmaxf(ixf, 0.0f), (float)(PSLEN - 1));
    const int iyc = (int)fminf(fmaxf(iyf, 0.0f), (float)(PSLEN - 1));
    const float v = tile[iyc * PSLEN + ixc];    // LDS gather (ds_load_b32)
    return valid ? v * wgt : 0.0f;
}

__global__ __launch_bounds__(BLOCK)
void tile_render_kernel(const float* __restrict__ locs,
                        const float* __restrict__ sources,
                        float* __restrict__ out) {
    const int b   = blockIdx.x;   // band   0..4
    const int p   = blockIdx.y;   // ptile  0..1023
    const int tid = threadIdx.x;

    __shared__ float smem[NSRC * NPIX];   // 43,264 B

    // ---- Stage the four source tiles for (p, b) into LDS -------------------
    // tile s lives at sources[((p*4 + s)*5 + b) * 2704 ...], contiguous 2704
    // floats, 16B-aligned (2704*4 % 16 == 0). smem layout matches: i*16 bytes.
    const int nvec = (NSRC * NPIX) / 4;   // 2704 float4 transfers
    for (int i = tid; i < nvec; i += BLOCK) {
        const int s = i / (NPIX / 4);
        const int j = i - s * (NPIX / 4);
        const float* g = sources +
            ((((size_t)p * NSRC + s) * NBAND + b) * NPIX) + (size_t)j * 4;
        float* l = smem + i * 4;
#if __has_builtin(__builtin_amdgcn_global_load_async_to_lds_b128)
        __builtin_amdgcn_global_load_async_to_lds_b128(
            (global_v4i*)g, (lds_v4i*)l, /*imm offset*/0, /*cpol*/0);
#else
        *(float4*)l = *(const float4*)g;
#endif
    }
#if __has_builtin(__builtin_amdgcn_s_wait_asynccnt)
    __builtin_amdgcn_s_wait_asynccnt(0);
#else
    asm volatile("s_wait_asynccnt 0" ::: "memory");
#endif
    __syncthreads();

    // ---- Per-source sub-pixel shifts (reference math, f32 order) -----------
    // l = (locs*(4/52) + 24/52 - 0.5) * 2 ; grid x uses l[...,1], y uses l[...,0]
    const float k  = (float)(4.0 / 52.0);
    const float c0 = (float)(24.0 / 52.0);
    float l0[NSRC], l1[NSRC];
#pragma unroll
    for (int s = 0; s < NSRC; ++s) {
        const float a0 = locs[((size_t)p * NSRC + s) * 2 + 0];
        const float a1 = locs[((size_t)p * NSRC + s) * 2 + 1];
        l0[s] = (a0 * k + c0 - 0.5f) * 2.0f;
        l1[s] = (a1 * k + c0 - 0.5f) * 2.0f;
    }

    // mgrid(i) = ((i - 25.5)/25.5) * (51/52)
    const float off    = (float)(PSLEN - 1) * 0.5f;   // 25.5
    const float mscale = (float)(51.0 / 52.0);

    // ---- Compute output pixels --------------------------------------------
    for (int pix = tid; pix < NPIX; pix += BLOCK) {
        const int h = pix / PSLEN;
        const int w = pix - h * PSLEN;
        const float cw = (((float)w - off) / off) * mscale;
        const float ch = (((float)h - off) / off) * mscale;

        float acc = 0.0f;
#pragma unroll
        for (int s = 0; s < NSRC; ++s) {
            const float ix  = ((cw - l1[s]) + 1.0f) * 0.5f * 51.0f;
            const float iy  = ((ch - l0[s]) + 1.0f) * 0.5f * 51.0f;
            const float ix0 = floorf(ix);
            const float iy0 = floorf(iy);
            const float wx1 = ix - ix0, wy1 = iy - iy0;
            const float wx0 = 1.0f - wx1, wy0 = 1.0f - wy1;
            const float* tile = smem + s * NPIX;
            acc += corner_tap(tile, iy0,        ix0,        wy0 * wx0);
            acc += corner_tap(tile, iy0,        ix0 + 1.0f, wy0 * wx1);
            acc += corner_tap(tile, iy0 + 1.0f, ix0,        wy1 * wx0);
            acc += corner_tap(tile, iy0 + 1.0f, ix0 + 1.0f, wy1 * wx1);
        }
        out[(((size_t)p * NBAND) + b) * NPIX + pix] = acc;   // coalesced
    }
}

extern "C" void kernel_launch(void* const* d_in, const int* in_sizes, int n_in,
                              void* d_out, int out_size, void* d_ws, size_t ws_size,
                              hipStream_t stream) {
    const float* locs    = (const float*)d_in[0];
    const float* sources = (const float*)d_in[1];
    float* out           = (float*)d_out;

    const int P = in_sizes[0] / (NSRC * 2);   // 1024 ptiles
    dim3 grid(NBAND, P);
    tile_render_kernel<<<grid, BLOCK, 0, stream>>>(locs, sources, out);
}